// ConvQuadInterp3d_37434934952187
// MI455X (gfx1250) — compile-verified
//
#include <hip/hip_runtime.h>
#include <stdint.h>

// Problem dimensions (fixed by the reference: x = (2,1,8,384,384) fp32).
#define BB 2
#define DD 8
#define HH 384
#define WW 384

// Tile: 32 (W) x 8 (H) per 256-thread block, 3 depth planes with halo in LDS.
#define TW 32
#define TH 8
#define LTW (TW + 2)   // 34
#define LTH (TH + 2)   // 10
#define TILE_ELEMS (3 * LTH * LTW)  // 1020 floats = 4080 B LDS

#define BONUS 10.0f
#define EPS 1e-7f

// ---- CDNA5 async global->LDS copy (gfx1250, ASYNCcnt-tracked) --------------
// LDS address comes from the vdst VGPR; generic shared-pointer low 32 bits == LDS
// byte offset (ISA flat-aperture rule), so truncating the generic address is
// exactly the value the instruction wants.
__device__ __forceinline__ void async_g2lds_b32(unsigned lds_off, const float* gsrc) {
    asm volatile("global_load_async_to_lds_b32 %0, %1, off"
                 :: "v"(lds_off), "v"(gsrc)
                 : "memory");
}
__device__ __forceinline__ void wait_asynccnt0() {
    asm volatile("s_wait_asynccnt 0" ::: "memory");
}

__global__ __launch_bounds__(TW * TH)
void quad_interp3d_kernel(const float* __restrict__ x,
                          const float* __restrict__ hes_noise,
                          float* __restrict__ coords,  // (B,3,D,H,W)
                          float* __restrict__ ymax)    // (B,D,H,W)
{
    __shared__ float tile[3][LTH][LTW];

    const int tx = threadIdx.x;
    const int ty = threadIdx.y;
    const int w0 = blockIdx.x * TW;
    const int h0 = blockIdx.y * TH;
    const int bz = blockIdx.z;           // b*DD + d
    const int b  = bz >> 3;
    const int d  = bz & (DD - 1);

    const int planeHW = HH * WW;
    const int volDHW  = DD * planeHW;
    const float* xb = x + (size_t)b * volDHW;

    // ---- Stage halo tile (3 x 10 x 34) via async global->LDS copies -------
    const int tid = ty * TW + tx;
    #pragma unroll
    for (int i = tid; i < TILE_ELEMS; i += TW * TH) {
        const int p  = i / (LTH * LTW);
        const int r  = i - p * (LTH * LTW);
        const int ly = r / LTW;
        const int lx = r - ly * LTW;
        int gd = d + p - 1;  gd = gd < 0 ? 0 : (gd > DD - 1 ? DD - 1 : gd);
        int gh = h0 + ly - 1; gh = gh < 0 ? 0 : (gh > HH - 1 ? HH - 1 : gh);
        int gw = w0 + lx - 1; gw = gw < 0 ? 0 : (gw > WW - 1 ? WW - 1 : gw);
        const float* src = xb + (size_t)gd * planeHW + (size_t)gh * WW + gw;
        const unsigned dst = (unsigned)(uintptr_t)&tile[p][ly][lx];
        async_g2lds_b32(dst, src);
    }
    wait_asynccnt0();
    __syncthreads();

    // ---- Noise matrix (uniform; 9 scalar-ish loads, L0-resident) -----------
    float nz[9];
    #pragma unroll
    for (int k = 0; k < 9; ++k) nz[k] = fabsf(hes_noise[k]) * EPS;

    const int h = h0 + ty;
    const int w = w0 + tx;
    const int lh = ty + 1;
    const int lw = tx + 1;

    // ---- Stencil reads from LDS -------------------------------------------
    #define T(P, DY, DX) tile[(P)][lh + (DY)][lw + (DX)]
    const float c  = T(1, 0, 0);
    const float wp = T(1, 0, 1),  wm = T(1, 0, -1);
    const float hp = T(1, 1, 0),  hm = T(1, -1, 0);
    const float dp = T(2, 0, 0),  dm = T(0, 0, 0);

    const float hmwm = T(1, -1, -1), hmwp = T(1, -1, 1);
    const float hpwm = T(1,  1, -1), hpwp = T(1,  1, 1);
    const float dmhm = T(0, -1, 0),  dmhp = T(0, 1, 0);
    const float dphm = T(2, -1, 0),  dphp = T(2, 1, 0);
    const float dmwm = T(0, 0, -1),  dmwp = T(0, 0, 1);
    const float dpwm = T(2, 0, -1),  dpwp = T(2, 0, 1);

    // 3x3x3 NMS max (clamped reads == windowed max: duplicates are in-window).
    float mx = -__builtin_inff();
    #pragma unroll
    for (int p = 0; p < 3; ++p)
        #pragma unroll
        for (int yy = -1; yy <= 1; ++yy)
            #pragma unroll
            for (int xx = -1; xx <= 1; ++xx)
                mx = fmaxf(mx, T(p, yy, xx));
    const bool mask = (c == mx);
    #undef T

    // ---- Derivatives -------------------------------------------------------
    const float gx = 0.5f * (wp - wm);
    const float gy = 0.5f * (hp - hm);
    const float gs = 0.5f * (dp - dm);
    const float dxx = wp - 2.0f * c + wm;
    const float dyy = hp - 2.0f * c + hm;
    const float dss = dp - 2.0f * c + dm;
    const float dxy = 0.25f * (hmwm - hmwp - hpwm + hpwp);
    const float dys = 0.25f * (dmhm - dmhp - dphm + dphp);
    const float dxs = 0.25f * (dmwm - dmwp - dpwm + dpwp);

    // ---- Masked 3x3 solve (general: noise makes Hessian asymmetric) --------
    float dx0 = 0.0f, dx1 = 0.0f, dx2 = 0.0f, dyv = 0.0f;
    if (mask) {
        const float a00 = dxx + nz[0], a01 = dxy + nz[1], a02 = dxs + nz[2];
        const float a10 = dxy + nz[3], a11 = dyy + nz[4], a12 = dys + nz[5];
        const float a20 = dxs + nz[6], a21 = dys + nz[7], a22 = dss + nz[8];
        const float b0 = gx, b1 = gy, b2 = gs;

        const float c00 = a11 * a22 - a12 * a21;
        const float c01 = a10 * a22 - a12 * a20;
        const float c02 = a10 * a21 - a11 * a20;
        const float det = a00 * c00 - a01 * c01 + a02 * c02;
        const float inv = 1.0f / det;

        const float s0 = (b0 * c00
                        - a01 * (b1 * a22 - a12 * b2)
                        + a02 * (b1 * a21 - a11 * b2)) * inv;
        const float s1 = (a00 * (b1 * a22 - a12 * b2)
                        - b0 * c01
                        + a02 * (a10 * b2 - b1 * a20)) * inv;
        const float s2 = (a00 * (a11 * b2 - b1 * a21)
                        - a01 * (a10 * b2 - b1 * a20)
                        + b0 * c02) * inv;

        dx0 = -s0; dx1 = -s1; dx2 = -s2;
        const float mabs = fmaxf(fabsf(dx0), fmaxf(fabsf(dx1), fabsf(dx2)));
        if (mabs > 0.7f) { dx0 = 0.0f; dx1 = 0.0f; dx2 = 0.0f; }
        dyv = 0.5f * (b0 * dx0 + b1 * dx1 + b2 * dx2);
    }

    const float yv = c + dyv + (mask ? BONUS : 0.0f);

    // ---- Stores (coalesced per plane) -------------------------------------
    const size_t voxel = (size_t)d * planeHW + (size_t)h * WW + w;
    ymax[(size_t)b * volDHW + voxel] = yv;

    // coords channel order after dx[:, ::-1]: (d + dx_s, h + dx_y, w + dx_x)
    float* cb = coords + (size_t)b * 3 * volDHW;
    cb[0 * (size_t)volDHW + voxel] = (float)d + dx2;
    cb[1 * (size_t)volDHW + voxel] = (float)h + dx1;
    cb[2 * (size_t)volDHW + voxel] = (float)w + dx0;
}

extern "C" void kernel_launch(void* const* d_in, const int* in_sizes, int n_in,
                              void* d_out, int out_size, void* d_ws, size_t ws_size,
                              hipStream_t stream) {
    (void)in_sizes; (void)n_in; (void)out_size; (void)d_ws; (void)ws_size;
    const float* x  = (const float*)d_in[0];
    const float* hn = (const float*)d_in[1];

    float* coords = (float*)d_out;                                // B*3*D*H*W
    float* ymax   = coords + (size_t)BB * 3 * DD * HH * WW;       // B*D*H*W

    dim3 grid(WW / TW, HH / TH, BB * DD);   // 12 x 48 x 16
    dim3 block(TW, TH, 1);                  // 256 threads = 8 wave32
    hipLaunchKernelGGL(quad_interp3d_kernel, grid, block, 0, stream,
                       x, hn, coords, ymax);
}